// GravityField_43679817400671
// MI455X (gfx1250) — compile-verified
//
#include <hip/hip_runtime.h>
#include <math.h>

typedef __attribute__((ext_vector_type(2))) float v2f;
typedef __attribute__((ext_vector_type(4))) float v4f;
typedef __attribute__((ext_vector_type(8))) float v8f;

#define DM 64
#define RF 64
#define NB 8
#define NT 4096
#define STRENGTH 0.1f
#define PHI_SCALE 0.17677669529663689f   /* sqrt(2/64) */

// ---------------------------------------------------------------------------
// Kernel 0: zero the phi_sum accumulator (B*R = 512 floats). Must be re-zeroed
// every call since kernel 1 accumulates with atomics.
// ---------------------------------------------------------------------------
__global__ void gf_zero_phisum(float* __restrict__ phisum) {
    int i = threadIdx.x;
    if (i < NB * RF) phisum[i] = 0.0f;
}

// ---------------------------------------------------------------------------
// Kernel 1: per-token features via f32 WMMA (v_wmma_f32_16x16x4_f32).
//   h   = relu(coords @ w1.T + b1)           (16x64 tile per wave)
//   mass= softplus(h @ w2.T + b2)            (cross-lane reduction)
//   phi = sqrt(2/R) * cos(coords @ W + b)    (16x64 tile per wave)
//   phi_sum[b,r] += sum_tokens phi*mass      (LDS atomics -> global atomics)
// Block = 256 threads = 8 waves, 16 tokens/wave, 128 tokens/block.
// ---------------------------------------------------------------------------
__global__ __launch_bounds__(256) void gf_features(
    const float* __restrict__ coords, const float* __restrict__ w1,
    const float* __restrict__ b1,     const float* __restrict__ w2,
    const float* __restrict__ b2,     const float* __restrict__ Wr,
    const float* __restrict__ brff,   float* __restrict__ phi,
    float* __restrict__ phisum)
{
    __shared__ float lsum[RF];
    const int tid  = threadIdx.x;
    const int lane = tid & 31;
    const int wave = tid >> 5;

    if (tid < RF) lsum[tid] = 0.0f;
    __syncthreads();

    const int tokBase = blockIdx.x * 128 + wave * 16;
    const int batch   = tokBase >> 12;          // T = 4096
    const int m       = lane & 15;              // row / col within 16-tile
    const int hi      = lane >> 4;              // half-wave select
    const int koff    = hi << 1;                // K pair: lanes0-15 -> 0, 16-31 -> 2

    const float* crow = coords + (size_t)(tokBase + m) * DM;

    v8f hacc[4]; v8f pacc[4];
    #pragma unroll
    for (int t = 0; t < 4; ++t) { hacc[t] = {}; pacc[t] = {}; }

    // K loop: 16 steps of K=4; one shared A-frag, 8 WMMAs per step.
    for (int k0 = 0; k0 < DM; k0 += 4) {
        v2f a;
        a.x = crow[k0 + koff];
        a.y = crow[k0 + koff + 1];
        #pragma unroll
        for (int t = 0; t < 4; ++t) {
            const int n = t * 16 + m;
            // B-frag for h = coords @ w1.T : B[k][n] = w1[n][k]
            v2f bw;
            bw.x = w1[n * DM + k0 + koff];
            bw.y = w1[n * DM + k0 + koff + 1];
            hacc[t] = __builtin_amdgcn_wmma_f32_16x16x4_f32(
                false, a, false, bw, (short)0, hacc[t], false, false);
            // B-frag for phi_pre = coords @ W : B[k][n] = W[k][n]
            v2f bp;
            bp.x = Wr[(k0 + koff) * RF + n];
            bp.y = Wr[(k0 + koff + 1) * RF + n];
            pacc[t] = __builtin_amdgcn_wmma_f32_16x16x4_f32(
                false, a, false, bp, (short)0, pacc[t], false, false);
        }
    }

    // ---- mass = softplus(relu(h + b1) @ w2.T + b2) ----
    // C/D layout: VGPR j holds row (j + 8*hi), col n = t*16 + (lane&15).
    float p[8];
    #pragma unroll
    for (int j = 0; j < 8; ++j) p[j] = 0.0f;
    #pragma unroll
    for (int t = 0; t < 4; ++t) {
        const int n   = t * 16 + m;
        const float wv = w2[n];
        const float bv = b1[n];
        #pragma unroll
        for (int j = 0; j < 8; ++j) {
            float h = hacc[t][j] + bv;
            h = h > 0.0f ? h : 0.0f;
            p[j] += h * wv;
        }
    }
    // reduce over the 16 lanes of each half (bit4 untouched by masks 8..1)
    #pragma unroll
    for (int s = 8; s >= 1; s >>= 1) {
        #pragma unroll
        for (int j = 0; j < 8; ++j) p[j] += __shfl_xor(p[j], s, 32);
    }
    const float b2v = b2[0];
    float mass[8];
    #pragma unroll
    for (int j = 0; j < 8; ++j) {
        const float x = p[j] + b2v;
        mass[j] = (x > 20.0f) ? x : log1pf(expf(x));   // softplus
    }

    // ---- phi = scale * cos(pacc + b); store + mass-weighted partial sum ----
    float psum[4];
    #pragma unroll
    for (int t = 0; t < 4; ++t) {
        const int n   = t * 16 + m;
        const float bv = brff[n];
        float s = 0.0f;
        #pragma unroll
        for (int j = 0; j < 8; ++j) {
            const int row = j + (hi << 3);
            const float ph = PHI_SCALE * __cosf(pacc[t][j] + bv);
            phi[(size_t)(tokBase + row) * RF + n] = ph;
            s += ph * mass[j];
        }
        psum[t] = s;
    }
    // fold the two halves (rows 0-7 and 8-15 share the same r), then LDS acc
    #pragma unroll
    for (int t = 0; t < 4; ++t) {
        const float s = psum[t] + __shfl_xor(psum[t], 16, 32);
        if (hi == 0) atomicAdd(&lsum[t * 16 + m], s);
    }
    __syncthreads();
    if (tid < RF) atomicAdd(&phisum[batch * RF + tid], lsum[tid]);
}

// ---------------------------------------------------------------------------
// Kernel 2: streaming pass.  One block per token (32768 blocks).
//   grav = 0.1 * sum_r phi[t,r]*phi_sum[b,r]
//   out  = G + grav on the diagonal
// 16 KB per token streamed as b128 with NON-TEMPORAL policy on both the G
// loads and the out stores: G/out are touched exactly once, and keeping this
// 1 GB stream out of the 192 MB L2 lets the 8 MB phi buffer (written by
// kernel 1, read here) stay L2-resident instead of round-tripping to HBM.
// ---------------------------------------------------------------------------
__global__ __launch_bounds__(256) void gf_apply(
    const float* __restrict__ G, const float* __restrict__ phi,
    const float* __restrict__ phisum, float* __restrict__ out)
{
    __shared__ float red[2];
    __shared__ float gsh;
    const int tok = blockIdx.x;
    const int b   = tok >> 12;
    const int tid = threadIdx.x;

    if (tid < 64) {
        float v = phi[(size_t)tok * RF + tid] * phisum[b * RF + tid];
        #pragma unroll
        for (int s = 16; s >= 1; s >>= 1) v += __shfl_xor(v, s, 32);
        if ((tid & 31) == 0) red[tid >> 5] = v;
    }
    __syncthreads();
    if (tid == 0) gsh = STRENGTH * (red[0] + red[1]);
    __syncthreads();
    const float g = gsh;

    const v4f* gi = (const v4f*)(G   + (size_t)tok * (DM * DM));
    v4f*       go = (v4f*)      (out + (size_t)tok * (DM * DM));

    #pragma unroll
    for (int it = 0; it < 4; ++it) {
        const int i = tid + it * 256;
        v4f v = __builtin_nontemporal_load(&gi[i]);   // global_load_b128, TH_NT
        const int flat = i << 2;
        if (((flat    ) % 65) == 0) v[0] += g;        // d1==d2 <=> flat % 65 == 0
        if (((flat + 1) % 65) == 0) v[1] += g;
        if (((flat + 2) % 65) == 0) v[2] += g;
        if (((flat + 3) % 65) == 0) v[3] += g;
        __builtin_nontemporal_store(v, &go[i]);       // global_store_b128, TH_NT
    }
}

// ---------------------------------------------------------------------------
extern "C" void kernel_launch(void* const* d_in, const int* in_sizes, int n_in,
                              void* d_out, int out_size, void* d_ws, size_t ws_size,
                              hipStream_t stream) {
    const float* G      = (const float*)d_in[0];
    const float* coords = (const float*)d_in[1];
    const float* w1     = (const float*)d_in[2];
    const float* b1     = (const float*)d_in[3];
    const float* w2     = (const float*)d_in[4];
    const float* b2     = (const float*)d_in[5];
    const float* Wr     = (const float*)d_in[6];
    const float* brff   = (const float*)d_in[7];
    float* out = (float*)d_out;

    // workspace: phi [B*T*R] floats (8 MB) then phi_sum [B*R] floats
    float* ws_phi    = (float*)d_ws;
    float* ws_phisum = ws_phi + (size_t)NB * NT * RF;

    gf_zero_phisum<<<1, 512, 0, stream>>>(ws_phisum);
    gf_features<<<(NB * NT) / 128, 256, 0, stream>>>(
        coords, w1, b1, w2, b2, Wr, brff, ws_phi, ws_phisum);
    gf_apply<<<NB * NT, 256, 0, stream>>>(G, ws_phi, ws_phisum, out);
}